// ToxGNN_7988639171339
// MI455X (gfx1250) — compile-verified
//
#include <hip/hip_runtime.h>

// ---------------------------------------------------------------------------
// ToxGNN for MI455X (gfx1250): 3x GCNConv + pool + MLP head.
// Dense GEMMs use V_WMMA_F32_16X16X4_F32 (fp32 matrix cores, wave32):
//   - K and NOUT compile-time -> fully unrolled, all-immediate addressing
//   - W pre-packed fragment-major -> one global_load_b64 per B fragment
//   - 4 M-tiles per wave share each B fragment (4 WMMAs per B load)
//   - no predicated loads; EXEC all-ones around every WMMA (ISA requirement)
// Message passing = float4 gathers (L2-resident h) + global_atomic_add_f32.
// ---------------------------------------------------------------------------

#define HID 64
#define NGRAPH 2048

typedef __attribute__((ext_vector_type(2))) float v2f;
typedef __attribute__((ext_vector_type(8))) float v8f;

// ---------------------------- utility kernels ------------------------------

__global__ void zero_f32_kernel(float* __restrict__ p, long long n) {
  long long t = blockIdx.x * (long long)blockDim.x + threadIdx.x;
  if (t < n) p[t] = 0.0f;
}

__global__ void deg_count_kernel(const int* __restrict__ dst,
                                 float* __restrict__ deg, int E) {
  int t = blockIdx.x * blockDim.x + threadIdx.x;
  if (t < E) atomicAdd(&deg[dst[t]], 1.0f);
}

__global__ void deg_finalize_kernel(float* __restrict__ deg,
                                    float* __restrict__ dinv, int N) {
  int t = blockIdx.x * blockDim.x + threadIdx.x;
  if (t < N) {
    float d = deg[t] + 1.0f;   // self-loop
    deg[t]  = d;
    dinv[t] = rsqrtf(d);
  }
}

// Pad x [N x MOLD] -> xpad [N x 16] (zero-filled tail columns).
__global__ void pad_x_kernel(const float* __restrict__ x,
                             float* __restrict__ xpad, int N, int MOLD) {
  long long t = blockIdx.x * (long long)blockDim.x + threadIdx.x;
  if (t >= (long long)N * 16) return;
  const int i = (int)(t >> 4);
  const int c = (int)(t & 15);
  xpad[t] = (c < MOLD) ? x[(size_t)i * MOLD + c] : 0.0f;
}

// Pack W [K x 64] row-major into WMMA-fragment-major layout:
//   wpk[((r>>1)*64 + n)*2 + (r&1)] = W[r*64 + n]
// so a lane's B pair (rows k+2*half, k+2*half+1 at column n) is contiguous
// -> one global_load_b64 per K-step with an immediate offset.
// Rows r in [Kpad) with r >= Krows are zero-filled (layer-1 padding).
__global__ void pack_w_kernel(const float* __restrict__ w,
                              float* __restrict__ wpk, int Krows, int Kpad) {
  int t = blockIdx.x * blockDim.x + threadIdx.x;
  if (t >= Kpad * HID) return;
  const int r = t >> 6;          // source row
  const int n = t & 63;          // column
  const float v = (r < Krows) ? w[t] : 0.0f;
  wpk[(((r >> 1) << 6) + n) * 2 + (r & 1)] = v;
}

// ------------------------------ WMMA GEMM ----------------------------------
// H[M x 64] = X[M x K] @ W[K x 64], fp32. One wave owns 4 vertically stacked
// 16x16 output tiles; the B fragment per K-step is shared by all 4 WMMAs.
// K is a compile-time multiple of 4, NOUT = 64 -> fully unrolled loop with
// immediate address offsets only. M is a multiple of 16; a 4-tile group may
// overhang mtiles: clamp row pointers (loads stay in-bounds, results of
// duplicate tiles never stored; store guard is wave-uniform so EXEC == ~0).
//
// V_WMMA_F32_16X16X4_F32 fragment layout (ISA 7.12.2):
//   A 16x4 : lanes 0-15 -> M=lane,K={0,1}; lanes 16-31 -> M=lane-16,K={2,3}
//   B 4x16 : v0 = B[k+2*half][n=lane%16], v1 = B[k+1+2*half][n]
//   C 16x16: vgpr r, lanes 0-15 -> (M=r, N=lane); lanes 16-31 -> (M=r+8, N)

template <int K>
__global__ void wmma_gemm_f32_kernel(const float* __restrict__ X,
                                     const float* __restrict__ Wpk,
                                     float* __restrict__ H, int M) {
  constexpr int NOUT   = HID;
  constexpr int NTILES = NOUT / 16;
  const int wave = blockIdx.x * (blockDim.x >> 5) + (threadIdx.x >> 5);
  const int lane = threadIdx.x & 31;
  const int mtiles = M >> 4;                 // M is a multiple of 16
  const int tg = wave / NTILES;              // group of 4 M-tiles
  const int tn = wave - tg * NTILES;
  if (tg * 4 >= mtiles) return;              // wave-uniform exit

  const int col0 = tn << 4;
  const int half = lane >> 4;                // 0: lanes 0-15, 1: lanes 16-31
  const int l    = lane & 15;

  // Clamped row pointers: overhanging tiles alias the last valid tile so
  // every load is in-bounds; their results are simply never stored.
  const float* Xa[4];
#pragma unroll
  for (int t = 0; t < 4; ++t) {
    int tile = tg * 4 + t;
    if (tile > mtiles - 1) tile = mtiles - 1;
    Xa[t] = X + ((size_t)(tile << 4) + l) * K + half * 2;
  }
  // Packed-B base: this lane's pair for step s sits at + s*256 floats.
  const float* Wb = Wpk + (size_t)((half << 6) + col0 + l) * 2;

  v8f c0 = {}, c1 = {}, c2 = {}, c3 = {};
#pragma unroll
  for (int s = 0; s < K / 4; ++s) {
    const v2f b  = *(const v2f*)(Wb + s * 256);      // global_load_b64, imm off
    const v2f a0 = *(const v2f*)(Xa[0] + s * 4);
    const v2f a1 = *(const v2f*)(Xa[1] + s * 4);
    const v2f a2 = *(const v2f*)(Xa[2] + s * 4);
    const v2f a3 = *(const v2f*)(Xa[3] + s * 4);
    c0 = __builtin_amdgcn_wmma_f32_16x16x4_f32(false, a0, false, b, (short)0, c0, false, false);
    c1 = __builtin_amdgcn_wmma_f32_16x16x4_f32(false, a1, false, b, (short)0, c1, false, false);
    c2 = __builtin_amdgcn_wmma_f32_16x16x4_f32(false, a2, false, b, (short)0, c2, false, false);
    c3 = __builtin_amdgcn_wmma_f32_16x16x4_f32(false, a3, false, b, (short)0, c3, false, false);
  }

  v8f acc[4] = {c0, c1, c2, c3};
#pragma unroll
  for (int t = 0; t < 4; ++t) {
    const int tile = tg * 4 + t;
    if (tile < mtiles) {                     // wave-uniform guard
      const int outr = (tile << 4) + half * 8;
      float* Hp = H + (size_t)outr * NOUT + col0 + l;
#pragma unroll
      for (int r = 0; r < 8; ++r) Hp[r * NOUT] = acc[t][r];
    }
  }
}

// --------------------------- message passing -------------------------------
// agg[dst] += dinv[src]*dinv[dst] * h[src]; one thread = one edge x 4 ch.

__global__ void edge_scatter_kernel(const float* __restrict__ h,
                                    const int* __restrict__ src,
                                    const int* __restrict__ dst,
                                    const float* __restrict__ dinv,
                                    float* __restrict__ agg, int E) {
  long long t = blockIdx.x * (long long)blockDim.x + threadIdx.x;
  if (t >= (long long)E * 16) return;
  const int e  = (int)(t >> 4);
  const int c4 = (int)(t & 15) << 2;
  const int s  = src[e];
  const int d  = dst[e];
  // Prefetch the next edge's source row (global_prefetch_b8); h is L2-resident.
  if (e + 1 < E) __builtin_prefetch(h + (size_t)src[e + 1] * HID + c4, 0, 1);
  const float w = dinv[s] * dinv[d];
  const float4 hv = *(const float4*)(h + (size_t)s * HID + c4);
  float* a = agg + (size_t)d * HID + c4;
  atomicAdd(a + 0, hv.x * w);
  atomicAdd(a + 1, hv.y * w);
  atomicAdd(a + 2, hv.z * w);
  atomicAdd(a + 3, hv.w * w);
}

// out = relu(agg + h/deg + b)
__global__ void layer_finalize_kernel(const float* __restrict__ agg,
                                      const float* __restrict__ h,
                                      const float* __restrict__ deg,
                                      const float* __restrict__ b,
                                      float* __restrict__ out, int N) {
  long long t = blockIdx.x * (long long)blockDim.x + threadIdx.x;
  if (t >= (long long)N * HID) return;
  const int i = (int)(t >> 6);
  const int c = (int)(t & 63);
  float v = agg[t] + h[t] / deg[i] + b[c];
  out[t] = v > 0.0f ? v : 0.0f;
}

// ------------------------------- pooling -----------------------------------

__global__ void pool_count_kernel(const int* __restrict__ batch,
                                  float* __restrict__ cnt, int N) {
  int t = blockIdx.x * blockDim.x + threadIdx.x;
  if (t < N) atomicAdd(&cnt[batch[t]], 1.0f);
}

// values are post-ReLU (>= 0): int-bit atomicMax == float max, 0-init is valid.
__global__ void pool_nodes_kernel(const float* __restrict__ x,
                                  const int* __restrict__ batch,
                                  float* __restrict__ psum,
                                  float* __restrict__ pmax, int N) {
  long long t = blockIdx.x * (long long)blockDim.x + threadIdx.x;
  if (t >= (long long)N * 16) return;
  const int i  = (int)(t >> 4);
  const int c4 = (int)(t & 15) << 2;
  const int g  = batch[i];
  const float4 v = *(const float4*)(x + (size_t)i * HID + c4);
  float* s = psum + (size_t)g * HID + c4;
  int*   m = (int*)(pmax + (size_t)g * HID + c4);
  atomicAdd(s + 0, v.x);  atomicMax(m + 0, __float_as_int(v.x));
  atomicAdd(s + 1, v.y);  atomicMax(m + 1, __float_as_int(v.y));
  atomicAdd(s + 2, v.z);  atomicMax(m + 2, __float_as_int(v.z));
  atomicAdd(s + 3, v.w);  atomicMax(m + 3, __float_as_int(v.w));
}

// gfeat[g, 0:64] = mean, gfeat[g, 64:128] = max
__global__ void gfeat_kernel(const float* __restrict__ psum,
                             const float* __restrict__ pmax,
                             const float* __restrict__ cnt,
                             float* __restrict__ gfeat, int G) {
  int t = blockIdx.x * blockDim.x + threadIdx.x;
  if (t >= G * 128) return;
  const int g = t >> 7;
  const int c = t & 127;
  float v;
  if (c < HID) {
    float n = cnt[g];
    v = psum[g * HID + c] / (n > 1.0f ? n : 1.0f);
  } else {
    v = pmax[g * HID + (c - HID)];
  }
  gfeat[t] = v;
}

// --------------------------------- head ------------------------------------

__global__ void relu_bias_kernel(float* __restrict__ h,
                                 const float* __restrict__ b, int G) {
  int t = blockIdx.x * blockDim.x + threadIdx.x;
  if (t >= G * HID) return;
  float v = h[t] + b[t & 63];
  h[t] = v > 0.0f ? v : 0.0f;
}

// out[g] = dot(hidden[g, 0:64], Wp2) + bp2   (one wave32 per graph)
__global__ void predict_out_kernel(const float* __restrict__ hidden,
                                   const float* __restrict__ Wp2,
                                   const float* __restrict__ bp2,
                                   float* __restrict__ out) {
  const int g = blockIdx.x;
  const int lane = threadIdx.x;
  float s = hidden[g * HID + lane]      * Wp2[lane] +
            hidden[g * HID + 32 + lane] * Wp2[32 + lane];
#pragma unroll
  for (int off = 16; off > 0; off >>= 1) s += __shfl_down(s, off, 32);
  if (lane == 0) out[g] = s + bp2[0];
}

// ------------------------------- launcher ----------------------------------

extern "C" void kernel_launch(void* const* d_in, const int* in_sizes, int n_in,
                              void* d_out, int out_size, void* d_ws, size_t ws_size,
                              hipStream_t stream) {
  const float* x     = (const float*)d_in[0];
  const int*   esrc  = (const int*)d_in[1];
  const int*   edst  = (const int*)d_in[2];
  const int*   batch = (const int*)d_in[3];
  const float* W1  = (const float*)d_in[4];
  const float* b1  = (const float*)d_in[5];
  const float* W2  = (const float*)d_in[6];
  const float* b2  = (const float*)d_in[7];
  const float* W3  = (const float*)d_in[8];
  const float* b3  = (const float*)d_in[9];
  const float* Wp1 = (const float*)d_in[10];
  const float* bp1 = (const float*)d_in[11];
  const float* Wp2 = (const float*)d_in[12];
  const float* bp2 = (const float*)d_in[13];

  const int N = in_sizes[3];          // nodes (len(batch))
  const int E = in_sizes[1];          // edges
  const int G = NGRAPH;
  const int MOLD = in_sizes[0] / N;   // 9

  float* ws     = (float*)d_ws;
  float* deg    = ws;                               // N
  float* dinv   = deg   + N;                        // N
  float* h      = dinv  + N;                        // N*64
  float* agg    = h     + (size_t)N * HID;          // N*64
  float* xcur   = agg   + (size_t)N * HID;          // N*64
  float* xpad   = xcur  + (size_t)N * HID;          // N*16
  float* W1pk   = xpad  + (size_t)N * 16;           // 16*64   (packed)
  float* W2pk   = W1pk  + 16 * HID;                 // 64*64   (packed)
  float* W3pk   = W2pk  + HID * HID;                // 64*64   (packed)
  float* Wp1pk  = W3pk  + HID * HID;                // 128*64  (packed)
  float* psum   = Wp1pk + 2 * HID * HID;            // G*64
  float* pmax   = psum  + (size_t)G * HID;          // G*64
  float* pcnt   = pmax  + (size_t)G * HID;          // G
  float* gfeat  = pcnt  + G;                        // G*128
  float* hidden = gfeat + (size_t)G * 128;          // G*64

  auto zero = [&](float* p, long long n) {
    int blocks = (int)((n + 255) / 256);
    zero_f32_kernel<<<blocks, 256, 0, stream>>>(p, n);
  };
  auto gemm_blocks = [&](int M) {
    long long waves = (long long)(((M >> 4) + 3) / 4) * (HID / 16);
    return (int)((waves + 7) / 8);                  // 8 wave32 per block
  };
  auto pack = [&](const float* w, float* wpk, int Krows, int Kpad) {
    pack_w_kernel<<<(Kpad * HID + 255) / 256, 256, 0, stream>>>(w, wpk, Krows, Kpad);
  };

  // degrees + symmetric normalization
  zero(deg, N);
  deg_count_kernel<<<(E + 255) / 256, 256, 0, stream>>>(edst, deg, E);
  deg_finalize_kernel<<<(N + 255) / 256, 256, 0, stream>>>(deg, dinv, N);

  // pad layer-1 features to K=16; pack all weights fragment-major
  pad_x_kernel<<<(int)(((long long)N * 16 + 255) / 256), 256, 0, stream>>>(
      x, xpad, N, MOLD);
  pack(W1,  W1pk,  MOLD,    16);
  pack(W2,  W2pk,  HID,     HID);
  pack(W3,  W3pk,  HID,     HID);
  pack(Wp1, Wp1pk, 2 * HID, 2 * HID);

  const long long edge_threads = (long long)E * 16;
  const long long node_threads = (long long)N * HID;
  const int eblocks = (int)((edge_threads + 255) / 256);
  const int nblocks = (int)((node_threads + 255) / 256);

  // ---- layer 1 (padded K=16) ----
  zero(agg, (long long)N * HID);
  wmma_gemm_f32_kernel<16><<<gemm_blocks(N), 256, 0, stream>>>(xpad, W1pk, h, N);
  edge_scatter_kernel<<<eblocks, 256, 0, stream>>>(h, esrc, edst, dinv, agg, E);
  layer_finalize_kernel<<<nblocks, 256, 0, stream>>>(agg, h, deg, b1, xcur, N);

  // ---- layer 2 ----
  zero(agg, (long long)N * HID);
  wmma_gemm_f32_kernel<HID><<<gemm_blocks(N), 256, 0, stream>>>(xcur, W2pk, h, N);
  edge_scatter_kernel<<<eblocks, 256, 0, stream>>>(h, esrc, edst, dinv, agg, E);
  layer_finalize_kernel<<<nblocks, 256, 0, stream>>>(agg, h, deg, b2, xcur, N);

  // ---- layer 3 ----
  zero(agg, (long long)N * HID);
  wmma_gemm_f32_kernel<HID><<<gemm_blocks(N), 256, 0, stream>>>(xcur, W3pk, h, N);
  edge_scatter_kernel<<<eblocks, 256, 0, stream>>>(h, esrc, edst, dinv, agg, E);
  layer_finalize_kernel<<<nblocks, 256, 0, stream>>>(agg, h, deg, b3, xcur, N);

  // ---- pooling ----
  zero(psum, (long long)G * HID);
  zero(pmax, (long long)G * HID);   // 0-init valid: xcur >= 0 post-ReLU
  zero(pcnt, G);
  pool_count_kernel<<<(N + 255) / 256, 256, 0, stream>>>(batch, pcnt, N);
  pool_nodes_kernel<<<(int)(((long long)N * 16 + 255) / 256), 256, 0, stream>>>(
      xcur, batch, psum, pmax, N);
  gfeat_kernel<<<(G * 128 + 255) / 256, 256, 0, stream>>>(psum, pmax, pcnt, gfeat, G);

  // ---- head: relu(g @ Wp1 + bp1) @ Wp2 + bp2 ----
  wmma_gemm_f32_kernel<2 * HID><<<gemm_blocks(G), 256, 0, stream>>>(
      gfeat, Wp1pk, hidden, G);
  relu_bias_kernel<<<(G * HID + 255) / 256, 256, 0, stream>>>(hidden, bp1, G);
  predict_out_kernel<<<G, 32, 0, stream>>>(hidden, Wp2, bp2, (float*)d_out);
}